// CodedNet_80444737454106
// MI455X (gfx1250) — compile-verified
//
#include <hip/hip_runtime.h>
#include <hip/hip_bf16.h>

#define P       256
#define CHN     28
#define JCHUNK  128
#define THREADS 128
#define NWAVE   (THREADS / 32)
#define CHUNK   (JCHUNK * CHN)        // 3584 floats per batch per WG
#define PIECE   (CHUNK / NWAVE)       // 896 floats per wave piece (16B multiple)

typedef unsigned int u32x4 __attribute__((ext_vector_type(4)));
typedef int          i32x8 __attribute__((ext_vector_type(8)));
typedef int          i32x4 __attribute__((ext_vector_type(4)));

#if __has_builtin(__builtin_amdgcn_tensor_load_to_lds)
#define HAVE_TDM 1
#else
#define HAVE_TDM 0
#endif

// LDS byte offset (workgroup-relative) of a __shared__ pointer.
__device__ __forceinline__ unsigned lds_off(const void* p) {
  return (unsigned)(unsigned long long)
      (__attribute__((address_space(3))) const char*)p;
}

#if HAVE_TDM
// 1-D TDM copy: nelem f32 from global gsrc -> LDS byte offset lds_byte.
// Tracked on TENSORcnt; caller waits with s_wait_tensorcnt.
__device__ __forceinline__ void tdm_load_1d(const float* gsrc,
                                            unsigned lds_byte,
                                            unsigned nelem) {
  unsigned long long ga = (unsigned long long)(const void*)gsrc;

  // D# group 0 (128b): count=1 | lds_addr | global_addr[56:0] | type=2
  u32x4 g0;
  g0[0] = 1u;                                          // count=1 (valid), user mode
  g0[1] = lds_byte;                                    // lds_addr
  g0[2] = (unsigned)(ga & 0xFFFFFFFFull);              // global_addr[31:0]
  g0[3] = ((unsigned)(ga >> 32) & 0x01FFFFFFu)         // global_addr[56:32]
        | (2u << 30);                                  // type = 2 ("image")

  // D# group 1 (256b): data_size=4B; tensor_dim0 = stride0 = nelem;
  // tile_dim0 = nelem; tile_dim1/2 = 0 (unused); no padding/iterate/multicast.
  i32x8 g1;
  g1[0] = (int)(2u << 16);                             // data_size=2 -> 4 bytes
  g1[1] = (int)((nelem & 0xFFFFu) << 16);              // tensor_dim0[15:0]
  g1[2] = (int)(((nelem >> 16) & 0xFFFFu)              // tensor_dim0[31:16]
        | (1u << 16));                                 // tensor_dim1 = 1
  g1[3] = (int)((nelem & 0xFFFFu) << 16);              // tile_dim0 = nelem
  g1[4] = 0;                                           // tile_dim1=0, tile_dim2=0
  g1[5] = (int)nelem;                                  // tensor_dim0_stride[31:0]
  g1[6] = 0;                                           // stride0 hi, stride1 lo
  g1[7] = 0;                                           // stride1 hi

  i32x4 gz = {0, 0, 0, 0};                             // groups 2/3: dims unused
#if defined(__clang_major__) && __clang_major__ >= 23
  i32x8 gz8 = {0, 0, 0, 0, 0, 0, 0, 0};
  __builtin_amdgcn_tensor_load_to_lds(g0, g1, gz, gz, gz8, 0);
#else
  __builtin_amdgcn_tensor_load_to_lds(g0, g1, gz, gz, 0);
#endif
}
#endif  // HAVE_TDM

__global__ __launch_bounds__(THREADS)
void CodedNet_80444737454106_kernel(const float* __restrict__ x,
                                    const float* __restrict__ bk,
                                    float* __restrict__ out, int B) {
  __shared__ __align__(16) float buf[2][CHUNK];        // 28,672 B double buffer

  const int tid = threadIdx.x;
  const int i   = blockIdx.y;                          // row
  const int j0  = blockIdx.x * JCHUNK;
  const int j   = j0 + tid;

  // ---- per-channel shifted mask into registers: m[ch] = bk[(i-ch)&255, j, ch]
  float m[CHN];
#pragma unroll
  for (int ch = 0; ch < CHN; ++ch) {
    const int is = (i - ch) & (P - 1);
    m[ch] = bk[((unsigned)(is * P + j)) * CHN + ch];
  }

  const unsigned long long bstride = (unsigned long long)P * P * CHN;
  const float* src0 =
      x + ((unsigned long long)i * P + j0) * CHN;

#if HAVE_TDM
  const int wid = __builtin_amdgcn_readfirstlane(tid >> 5);  // wave id, uniform
  const float* wsrc = src0 + (unsigned)wid * PIECE;
  const unsigned lds0 = lds_off(&buf[0][0]) + (unsigned)wid * (PIECE * 4u);
  const unsigned lds1 = lds_off(&buf[1][0]) + (unsigned)wid * (PIECE * 4u);

  // prefetch b = 0
  tdm_load_1d(wsrc, lds0, PIECE);
#endif

  for (int b = 0; b < B; ++b) {
    const int cur = b & 1;

#if HAVE_TDM
    __builtin_amdgcn_s_wait_tensorcnt(0);   // this wave's piece of buf[cur] done
    __syncthreads();                        // all pieces done; prev reads retired
    if (b + 1 < B) {                        // prefetch b+1 into other buffer
      tdm_load_1d(wsrc + (unsigned long long)(b + 1) * bstride,
                  (cur ? lds0 : lds1), PIECE);
    }
    const float4* v =
        (const float4*)&buf[cur][(unsigned)tid * CHN];   // 112B/lane, 16B aligned
#else
    const float4* v =
        (const float4*)(src0 + (unsigned long long)b * bstride
                        + (unsigned)tid * CHN);
#endif

    float acc = 0.0f;
#pragma unroll
    for (int k = 0; k < CHN / 4; ++k) {
      const float4 q = v[k];
      acc = __builtin_fmaf(q.x, m[4 * k + 0], acc);
      acc = __builtin_fmaf(q.y, m[4 * k + 1], acc);
      acc = __builtin_fmaf(q.z, m[4 * k + 2], acc);
      acc = __builtin_fmaf(q.w, m[4 * k + 3], acc);
    }

    out[((unsigned long long)b * P + i) * P + j] = acc;
  }
}

extern "C" void kernel_launch(void* const* d_in, const int* in_sizes, int n_in,
                              void* d_out, int out_size, void* d_ws,
                              size_t ws_size, hipStream_t stream) {
  const float* x  = (const float*)d_in[0];   // [B, 256, 256, 28] f32
  const float* bk = (const float*)d_in[1];   // [256, 256, 28] f32
  float* out      = (float*)d_out;           // [B, 256, 256] f32

  const int B = in_sizes[0] / (P * P * CHN); // = 32 for the reference

  dim3 grid(P / JCHUNK, P, 1);               // (2, 256)
  CodedNet_80444737454106_kernel<<<grid, dim3(THREADS), 0, stream>>>(x, bk, out,
                                                                     B);
}